// Hoppooling_mix_4544075399265
// MI455X (gfx1250) — compile-verified
//
#include <hip/hip_runtime.h>
#include <hip/hip_bf16.h>
#include <stdint.h>

// ---------------- problem constants (fixed by the reference) ----------------
constexpr int NUM_GRAPHS = 128;
constexpr int NPG        = 1024;                 // nodes per graph
constexpr int N_NODES    = NUM_GRAPHS * NPG;     // 131072
constexpr int AVG_DEG    = 32;
constexpr int N_EDGES    = N_NODES * AVG_DEG;    // 4194304
constexpr int D_FEAT     = 128;
constexpr int K_KEEP     = 512;                  // ceil(0.5 * 1024)

typedef __attribute__((ext_vector_type(2))) float v2f;
typedef __attribute__((ext_vector_type(8))) float v8f;

// ---------------------------------------------------------------------------
// zero three node-sized arrays (workspace must be re-zeroed every call)
__global__ void k_zero3(float* __restrict__ ha, float* __restrict__ hb,
                        float* __restrict__ sc) {
  int i = blockIdx.x * blockDim.x + threadIdx.x;
  if (i < N_NODES) { ha[i] = 0.0f; hb[i] = 0.0f; sc[i] = 0.0f; }
}

__global__ void k_zero1(float* __restrict__ h) {
  int i = blockIdx.x * blockDim.x + threadIdx.x;
  if (i < N_NODES) h[i] = 0.0f;
}

// hop 1: h[src[e]] += w[e]
__global__ void k_scatter_w(const int* __restrict__ src,
                            const float* __restrict__ w,
                            float* __restrict__ h) {
  int e = blockIdx.x * blockDim.x + threadIdx.x;
  if (e < N_EDGES) {
    __builtin_prefetch(src + e + 8192, 0, 1);   // global_prefetch_b8
    __hip_atomic_fetch_add(&h[src[e]], w[e], __ATOMIC_RELAXED,
                           __HIP_MEMORY_SCOPE_AGENT);
  }
}

// hops 2..: hout[src[e]] += hin[dst[e]] * w[e]
__global__ void k_scatter_hw(const int* __restrict__ src,
                             const int* __restrict__ dst,
                             const float* __restrict__ w,
                             const float* __restrict__ hin,
                             float* __restrict__ hout) {
  int e = blockIdx.x * blockDim.x + threadIdx.x;
  if (e < N_EDGES) {
    __builtin_prefetch(src + e + 8192, 0, 1);
    __builtin_prefetch(dst + e + 8192, 0, 1);
    float v = hin[dst[e]] * w[e];
    __hip_atomic_fetch_add(&hout[src[e]], v, __ATOMIC_RELAXED,
                           __HIP_MEMORY_SCOPE_AGENT);
  }
}

__global__ void k_addscore(float* __restrict__ sc, const float* __restrict__ h) {
  int i = blockIdx.x * blockDim.x + threadIdx.x;
  if (i < N_NODES) sc[i] += h[i];
}

// ---------------------------------------------------------------------------
// per-graph stable top-k via bitonic sort of u64 keys in LDS.
// key = (~float_bits(score) << 32) | local_idx  (scores >= 0), ascending sort
// => descending score, ascending index on ties (matches lax.top_k).
// grid = NUM_GRAPHS blocks, block = 1024 threads (32 wave32 per WGP max).
__global__ void k_topk(const float* __restrict__ score,
                       float* __restrict__ mask,
                       float* __restrict__ permOut) {
  __shared__ unsigned long long s[NPG];
  const int g = blockIdx.x;
  const int i = threadIdx.x;

  float sv = score[g * NPG + i];
  unsigned int bits = __float_as_uint(sv);
  s[i] = ((unsigned long long)(~bits) << 32) | (unsigned int)i;
  __syncthreads();

  for (int k = 2; k <= NPG; k <<= 1) {
    for (int j = k >> 1; j > 0; j >>= 1) {
      int ixj = i ^ j;
      if (ixj > i) {
        unsigned long long a = s[i], b = s[ixj];
        bool up = ((i & k) == 0);
        if ((a > b) == up) { s[i] = b; s[ixj] = a; }
      }
      __syncthreads();
    }
  }

  unsigned long long key = s[i];
  int local = (int)(unsigned int)(key & 0xFFFFFFFFull);
  int node  = g * NPG + local;
  mask[node] = (i < K_KEEP) ? 1.0f : 0.0f;
  if (i < K_KEEP) permOut[g * K_KEEP + i] = (float)node;
}

// ---------------------------------------------------------------------------
// x2 = diag(mask) * X, exact in f32 via V_WMMA_F32_16X16X4_F32.
// One wave per 16(node)x16(feat) tile; 4 WMMAs (K=4 slabs) per tile.
//   A_c[m,k] = (m == 4c+k) ? mask[nb+m] : 0     (A: lane=M, halfwave/VGPR=K)
//   B_c[k,n] = x[(nb+4c+k)*D + fb+n]            (B: lane=N, halfwave/VGPR=K)
//   D[m,n]   = mask[nb+m] * x[(nb+m)*D + fb+n]  (C/D: VGPR r -> M=r+8*hi, N=lane&15)
__global__ void k_x2_wmma(const float* __restrict__ x,
                          const float* __restrict__ mask,
                          float* __restrict__ x2) {
  const int lane = threadIdx.x & 31;
  const int wid  = threadIdx.x >> 5;
  const int tile = blockIdx.x * (blockDim.x >> 5) + wid;  // exactly 65536 tiles
  const int nt = tile >> 3;          // node tile (8192)
  const int ft = tile & 7;           // feat tile (8)
  const int nb = nt * 16;
  const int fb = ft * 16;
  const int n16 = lane & 15;
  const int hi  = lane >> 4;         // half-wave select
  const int k0  = hi * 2;            // K index of v[0] in A/B slabs

  const float mval = mask[nb + n16]; // A-side: m = n16

  v8f acc = {};
#pragma unroll
  for (int c = 0; c < 4; ++c) {
    v2f A, B;
    A[0] = (n16 == 4 * c + k0)     ? mval : 0.0f;
    A[1] = (n16 == 4 * c + k0 + 1) ? mval : 0.0f;
    B[0] = x[(size_t)(nb + 4 * c + k0)     * D_FEAT + fb + n16];
    B[1] = x[(size_t)(nb + 4 * c + k0 + 1) * D_FEAT + fb + n16];
    acc = __builtin_amdgcn_wmma_f32_16x16x4_f32(
        /*neg_a=*/false, A, /*neg_b=*/false, B,
        /*c_mod=*/(short)0, acc, /*reuse_a=*/false, /*reuse_b=*/false);
  }

  const int nodeHi = nb + hi * 8;
#pragma unroll
  for (int r = 0; r < 8; ++r)
    x2[(size_t)(nodeHi + r) * D_FEAT + fb + n16] = acc[r];
}

// ew2[e] = w[e] * mask[src[e]] * mask[dst[e]]   (mask in {0,1})
__global__ void k_edgemask(const int* __restrict__ src,
                           const int* __restrict__ dst,
                           const float* __restrict__ w,
                           const float* __restrict__ mask,
                           float* __restrict__ ew2) {
  int e = blockIdx.x * blockDim.x + threadIdx.x;
  if (e < N_EDGES) {
    __builtin_prefetch(src + e + 8192, 0, 1);
    __builtin_prefetch(dst + e + 8192, 0, 1);
    ew2[e] = w[e] * mask[src[e]] * mask[dst[e]];
  }
}

// ---------------------------------------------------------------------------
extern "C" void kernel_launch(void* const* d_in, const int* in_sizes, int n_in,
                              void* d_out, int out_size, void* d_ws, size_t ws_size,
                              hipStream_t stream) {
  const float* x   = (const float*)d_in[0];
  const int*   ei  = (const int*)d_in[1];
  const float* w   = (const float*)d_in[2];
  // d_in[3] = batch (unused: graphs are equal-sized blocks)
  const int* src = ei;
  const int* dst = ei + N_EDGES;

  float* out     = (float*)d_out;
  float* x2      = out;                                  // N_NODES * D_FEAT
  float* ew2     = out + (size_t)N_NODES * D_FEAT;       // N_EDGES
  float* permOut = ew2 + N_EDGES;                        // NUM_GRAPHS * K_KEEP

  float* ws = (float*)d_ws;                              // 4 * N_NODES floats (2 MB)
  float* ha = ws;
  float* hb = ws + N_NODES;
  float* sc = ws + 2 * N_NODES;
  float* mk = ws + 3 * N_NODES;

  const int TB = 256;
  const int nodeBlocks = (N_NODES + TB - 1) / TB;   // 512
  const int edgeBlocks = (N_EDGES + TB - 1) / TB;   // 16384

  // hop 0: zero workspace, h1 = segsum(w, src), score = h1
  k_zero3<<<nodeBlocks, TB, 0, stream>>>(ha, hb, sc);
  k_scatter_w<<<edgeBlocks, TB, 0, stream>>>(src, w, ha);
  k_addscore<<<nodeBlocks, TB, 0, stream>>>(sc, ha);

  // hop 1: h2 = segsum(h1[dst]*w, src), score += h2
  k_scatter_hw<<<edgeBlocks, TB, 0, stream>>>(src, dst, w, ha, hb);
  k_addscore<<<nodeBlocks, TB, 0, stream>>>(sc, hb);

  // hop 2: h3 = segsum(h2[dst]*w, src), score += h3
  k_zero1<<<nodeBlocks, TB, 0, stream>>>(ha);
  k_scatter_hw<<<edgeBlocks, TB, 0, stream>>>(src, dst, w, hb, ha);
  k_addscore<<<nodeBlocks, TB, 0, stream>>>(sc, ha);

  // per-graph stable top-k -> perm (+ node mask)
  k_topk<<<NUM_GRAPHS, NPG, 0, stream>>>(sc, mk, permOut);

  // x2 = diag(mask) * x via f32 WMMA; 65536 tiles, 8 waves/block
  k_x2_wmma<<<(N_NODES / 16) * (D_FEAT / 16) / 8, TB, 0, stream>>>(x, mk, x2);

  // edge_weight2 = w * mask[src] * mask[dst]
  k_edgemask<<<edgeBlocks, TB, 0, stream>>>(src, dst, w, mk, ew2);
}